// Batched_STHD_SpGAT_75814762709189
// MI455X (gfx1250) — compile-verified
//
#include <hip/hip_runtime.h>
#include <hip/hip_bf16.h>
#include <math.h>

#define NSUB   10000
#define NCLS   17
#define NG     500
#define NE     80000
#define HIDN   8
#define NTILES (NSUB / 16)   // 625 = 5^4

// GEMM kernel blocking: 5 waves per block, each wave owns a 16-row tile.
#define WPB    5
#define TPB    (WPB * 32)    // 160 threads
#define NBLK   (NTILES / WPB) // 125 blocks
#define KCHUNK 100           // K-slice staged in LDS (32KB per chunk)
#define NCHUNK (NG / KCHUNK) // 5

typedef __attribute__((ext_vector_type(2))) float v2f;
typedef __attribute__((ext_vector_type(8))) float v8f;

// ---------------- workspace layout (float offsets) ----------------
#define OFF_LOGP   0                         // [NSUB*17]
#define OFF_S      (OFF_LOGP + NSUB*NCLS)    // [NSUB]
#define OFF_XLR    (OFF_S + NSUB)            // [NSUB*16]  cols 0..7 = xl, 8..15 = xr
#define OFF_EATT   (OFF_XLR + NSUB*16)       // [NE]
#define OFF_EX     (OFF_EATT + NE)           // [NE]
#define OFF_M      (OFF_EX + NE)             // [NSUB] segment max
#define OFF_DEN    (OFF_M + NSUB)            // [NSUB] segment sum
#define OFF_IVT    (OFF_DEN + NSUB)          // [NG*32] 1/Var, pair-interleaved, padded to 32 classes
#define OFF_MVT    (OFF_IVT + NG*32)         // [NG*32] Mu/Var, pair-interleaved
#define OFF_WLR    (OFF_MVT + NG*32)         // [NG*16] [W_l|W_r], pair-interleaved
#define OFF_DC     (OFF_WLR + NG*16)         // [32]  D[c] = sum Mu^2/Var
#define OFF_BLR    (OFF_DC + 32)             // [16]  [b_l|b_r]

// ---------------- init: zero scalars, -inf segment max ----------------
__global__ void k_init(float* out, float* m, float* den) {
    int t = blockIdx.x * blockDim.x + threadIdx.x;
    if (t < NSUB) { m[t] = __int_as_float(0xFF800000); den[t] = 0.f; }
    if (t < 2) out[t] = 0.f;
}

// ---------------- class constants + packed B panels ----------------
__global__ void k_const(const float* __restrict__ Mu, const float* __restrict__ Var,
                        const float* __restrict__ Wl, const float* __restrict__ bl,
                        const float* __restrict__ Wr, const float* __restrict__ br,
                        float* IVt, float* MVt, float* WLR, float* Dc, float* blr) {
    int t = blockIdx.x * blockDim.x + threadIdx.x;
    if (t < NG * 32) {
        int k = t >> 5, c = t & 31;
        float iv = 0.f, mv = 0.f;
        if (c < NCLS) { float v = Var[c * NG + k]; iv = 1.f / v; mv = Mu[c * NG + k] * iv; }
        // pair-interleave so a lane reads (k,c),(k+1,c) as one b64
        int pk = (k >> 1) * 64 + c * 2 + (k & 1);
        IVt[pk] = iv;
        MVt[pk] = mv;
        if (c < 16) {
            float wv = (c < HIDN) ? Wl[k * HIDN + c] : Wr[k * HIDN + (c - HIDN)];
            WLR[(k >> 1) * 32 + c * 2 + (k & 1)] = wv;
        }
    }
    if (t < NCLS) {
        float acc = 0.f;
        for (int k = 0; k < NG; ++k) { float mu = Mu[t * NG + k]; acc += mu * mu / Var[t * NG + k]; }
        Dc[t] = acc;
    }
    if (t < 32 && t >= NCLS) Dc[t] = 0.f;
    if (t < 16) blr[t] = (t < HIDN) ? bl[t] : br[t - HIDN];
}

// ---------------- per-node softmax, logP, gather s ----------------
__global__ void k_softmax(const int* __restrict__ subset, const float* __restrict__ Wp,
                          const float* __restrict__ Sp, float* __restrict__ P,
                          float* __restrict__ logP, float* __restrict__ s) {
    int n = blockIdx.x * blockDim.x + threadIdx.x;
    if (n >= NSUB) return;
    int idx = subset[n];
    float w[NCLS], mx = -3.402823e38f;
    #pragma unroll
    for (int c = 0; c < NCLS; ++c) { w[c] = Wp[idx * NCLS + c]; mx = fmaxf(mx, w[c]); }
    float sum = 0.f;
    #pragma unroll
    for (int c = 0; c < NCLS; ++c) { float e = __expf(w[c] - mx); w[c] = e; sum += e; }
    float inv = 1.f / sum;
    #pragma unroll
    for (int c = 0; c < NCLS; ++c) {
        float p = w[c] * inv;
        P[n * NCLS + c] = p;
        logP[n * NCLS + c] = __logf(p + 1e-8f);
    }
    s[n] = Sp[idx];
}

// ---------------- fused triple-GEMM over x, LDS-staged B panels ----------------
// accIV = x^2 @ (1/Var)^T (2 N-tiles), accMV = x @ (Mu/Var)^T (2 N-tiles),
// accLR = x @ [W_l|W_r].  K = 500 via V_WMMA_F32_16X16X4_F32, panels shared by
// 5 waves through LDS in 32KB chunks.
__global__ __launch_bounds__(TPB) void k_gemm_fused(
        const float* __restrict__ x, const float* __restrict__ IVt,
        const float* __restrict__ MVt, const float* __restrict__ WLR,
        const float* __restrict__ Dc, const float* __restrict__ blr,
        const float* __restrict__ s, const float* __restrict__ P,
        float* __restrict__ xlr, float* __restrict__ out) {
    __shared__ __align__(16) float IV_s[KCHUNK * 32];   // 12.8 KB
    __shared__ __align__(16) float MV_s[KCHUNK * 32];   // 12.8 KB
    __shared__ __align__(16) float WL_s[KCHUNK * 16];   //  6.4 KB

    const int t     = threadIdx.x;
    const int wave  = t >> 5;
    const int l     = t & 31;
    const int tile  = blockIdx.x * WPB + wave;   // 0..624
    const int row0  = tile * 16;
    const int m16   = l & 15;
    const int khalf = l >> 4;                    // which K-pair of the 4-wide step

    v8f accIV0 = {}, accIV1 = {}, accMV0 = {}, accMV1 = {}, accLR = {};

    const float* xrow = x + (size_t)(row0 + m16) * NG;

    for (int ch = 0; ch < NCHUNK; ++ch) {
        const int kbase = ch * KCHUNK;
        // ---- cooperative stage of this K-slice of the panels (contiguous in
        //      the pair-interleaved layout): 800 + 800 + 400 float4 ----
        {
            const float4* gIV = (const float4*)(IVt + kbase * 32);
            const float4* gMV = (const float4*)(MVt + kbase * 32);
            const float4* gWL = (const float4*)(WLR + kbase * 16);
            float4* sIV = (float4*)IV_s;
            float4* sMV = (float4*)MV_s;
            float4* sWL = (float4*)WL_s;
            for (int i = t; i < 2000; i += TPB) {
                if (i < 800)       sIV[i]        = gIV[i];
                else if (i < 1600) sMV[i - 800]  = gMV[i - 800];
                else               sWL[i - 1600] = gWL[i - 1600];
            }
        }
        __syncthreads();

        for (int k0 = 0; k0 < KCHUNK; k0 += 4) {
            int kk  = k0 + 2 * khalf;                  // even -> 8B aligned
            v2f ax  = *(const v2f*)(xrow + kbase + kk);
            v2f ax2 = ax * ax;
            int pb  = (kk >> 1) * 64;                  // pair-interleaved row
            v2f bIV0 = *(const v2f*)(IV_s + pb + m16 * 2);
            v2f bIV1 = *(const v2f*)(IV_s + pb + (16 + m16) * 2);
            v2f bMV0 = *(const v2f*)(MV_s + pb + m16 * 2);
            v2f bMV1 = *(const v2f*)(MV_s + pb + (16 + m16) * 2);
            v2f bLR  = *(const v2f*)(WL_s + (kk >> 1) * 32 + m16 * 2);

            accIV0 = __builtin_amdgcn_wmma_f32_16x16x4_f32(false, ax2, false, bIV0, (short)0, accIV0, false, false);
            accIV1 = __builtin_amdgcn_wmma_f32_16x16x4_f32(false, ax2, false, bIV1, (short)0, accIV1, false, false);
            accMV0 = __builtin_amdgcn_wmma_f32_16x16x4_f32(false, ax,  false, bMV0, (short)0, accMV0, false, false);
            accMV1 = __builtin_amdgcn_wmma_f32_16x16x4_f32(false, ax,  false, bMV1, (short)0, accMV1, false, false);
            accLR  = __builtin_amdgcn_wmma_f32_16x16x4_f32(false, ax,  false, bLR,  (short)0, accLR,  false, false);
        }
        __syncthreads();
    }

    // C/D layout: VGPR r, lanes 0-15 -> M=r, lanes 16-31 -> M=r+8; N = l&15
    float d0  = Dc[m16];
    float d16 = Dc[16];
    float bj  = blr[m16];
    float llpart = 0.f;
    #pragma unroll
    for (int r = 0; r < 8; ++r) {
        int row = row0 + r + 8 * khalf;
        float sv = s[row];
        // classes 0..15 (N-tile 0)
        float F0 = -0.5f * (accIV0[r] - 2.f * sv * accMV0[r] + sv * sv * d0);
        llpart += P[row * NCLS + m16] * F0;
        // class 16 lives in column 0 of N-tile 1; columns 1..15 are zero padding
        if (m16 == 0) {
            float F1 = -0.5f * (accIV1[r] - 2.f * sv * accMV1[r] + sv * sv * d16);
            llpart += P[row * NCLS + 16] * F1;
        }
        // GAT node transforms with bias
        xlr[row * 16 + m16] = accLR[r] + bj;
    }
    // wave32 reduction + one atomic per tile
    #pragma unroll
    for (int off = 16; off > 0; off >>= 1) llpart += __shfl_xor(llpart, off, 32);
    if (l == 0) atomicAdd(&out[0], llpart * (1.f / (float)NSUB));
}

// ---------------- edge passes ----------------
__device__ inline void atomicMaxF(float* addr, float v) {
    if (v >= 0.f) atomicMax((int*)addr, __float_as_int(v));
    else          atomicMin((unsigned int*)addr, (unsigned int)__float_as_int(v));
}

__global__ void k_edge_score(const int* __restrict__ src, const int* __restrict__ dst,
                             const float* __restrict__ xlr, const float* __restrict__ att,
                             float* __restrict__ eatt, float* __restrict__ m) {
    int e = blockIdx.x * blockDim.x + threadIdx.x;
    if (e >= NE) return;
    int si = src[e], di = dst[e];
    const float* xl = xlr + si * 16;       // cols 0..7
    const float* xr = xlr + di * 16 + 8;   // cols 8..15
    float acc = 0.f;
    #pragma unroll
    for (int j = 0; j < HIDN; ++j) {
        float h  = xl[j] + xr[j];
        float lr = (h > 0.f) ? h : 0.2f * h;
        acc += lr * att[j];
    }
    eatt[e] = acc;
    atomicMaxF(&m[di], acc);
}

__global__ void k_edge_exp(const int* __restrict__ dst, const float* __restrict__ eatt,
                           const float* __restrict__ m, float* __restrict__ ex,
                           float* __restrict__ den) {
    int e = blockIdx.x * blockDim.x + threadIdx.x;
    if (e >= NE) return;
    int di  = dst[e];
    float v = __expf(eatt[e] - m[di]);
    ex[e]   = v;
    atomicAdd(&den[di], v);
}

__global__ void k_edge_ce(const int* __restrict__ src, const int* __restrict__ dst,
                          const float* __restrict__ ex, const float* __restrict__ den,
                          const float* __restrict__ P, const float* __restrict__ logP,
                          float* __restrict__ out) {
    int e = blockIdx.x * blockDim.x + threadIdx.x;
    if (e >= NE) return;
    int si = src[e], di = dst[e];
    float alpha = ex[e] / den[di];
    float dot = 0.f;
    #pragma unroll
    for (int c = 0; c < NCLS; ++c) dot += P[si * NCLS + c] * logP[di * NCLS + c];
    atomicAdd(&out[1], -(alpha * dot) * (1.f / (float)NSUB));
}

// ---------------- launch ----------------
extern "C" void kernel_launch(void* const* d_in, const int* in_sizes, int n_in,
                              void* d_out, int out_size, void* d_ws, size_t ws_size,
                              hipStream_t stream) {
    const float* x      = (const float*)d_in[0];
    const float* Mu     = (const float*)d_in[1];
    const float* Var    = (const float*)d_in[2];
    const int*   ei     = (const int*)d_in[3];   // [2, NE]: src row then dst row
    const int*   subset = (const int*)d_in[4];
    const float* Wp     = (const float*)d_in[5];
    const float* Sp     = (const float*)d_in[6];
    const float* Wl     = (const float*)d_in[7];
    const float* bl     = (const float*)d_in[8];
    const float* Wr     = (const float*)d_in[9];
    const float* br     = (const float*)d_in[10];
    const float* att    = (const float*)d_in[11];

    float* out  = (float*)d_out;         // [0]=ll_prot [1]=ce_space [2..]=P_sub
    float* P    = out + 2;
    float* w    = (float*)d_ws;
    float* logP = w + OFF_LOGP;
    float* s    = w + OFF_S;
    float* xlr  = w + OFF_XLR;
    float* eatt = w + OFF_EATT;
    float* ex   = w + OFF_EX;
    float* m    = w + OFF_M;
    float* den  = w + OFF_DEN;
    float* IVt  = w + OFF_IVT;
    float* MVt  = w + OFF_MVT;
    float* WLR  = w + OFF_WLR;
    float* Dc   = w + OFF_DC;
    float* blr  = w + OFF_BLR;

    const int* src = ei;
    const int* dst = ei + NE;

    k_init<<<(NSUB + 255) / 256, 256, 0, stream>>>(out, m, den);
    k_const<<<(NG * 32 + 255) / 256, 256, 0, stream>>>(Mu, Var, Wl, bl, Wr, br,
                                                       IVt, MVt, WLR, Dc, blr);
    k_softmax<<<(NSUB + 255) / 256, 256, 0, stream>>>(subset, Wp, Sp, P, logP, s);
    k_gemm_fused<<<NBLK, TPB, 0, stream>>>(x, IVt, MVt, WLR, Dc, blr, s, P, xlr, out);
    k_edge_score<<<(NE + 255) / 256, 256, 0, stream>>>(src, dst, xlr, att, eatt, m);
    k_edge_exp<<<(NE + 255) / 256, 256, 0, stream>>>(dst, eatt, m, ex, den);
    k_edge_ce<<<(NE + 255) / 256, 256, 0, stream>>>(src, dst, ex, den, P, logP, out);
}